// ObjectDetector_30451318129100
// MI455X (gfx1250) — compile-verified
//
#include <hip/hip_runtime.h>
#include <hip/hip_bf16.h>
#include <stdint.h>

// Problem constants: hmap [1,512,512,80] fp32, rreg [1,512,512,160], bbox [1,512,512,2]
#define HH 512
#define WW 512
#define CC 80
#define TX 16          // x-positions per tile
#define JJ (TX + 2)    // tile columns incl. halo (18)
#define TILES 16       // tiles per block (half a row)
#define LANELOADS (3 * JJ * (CC / 4))   // 1080 b128 lane-transfers per tile
#define LPT 5          // padded async-load instructions per thread per tile
#define NBIN 4096
#define TOPK 128
#define THREADS 256
#define CAND_MAX 65536u
#define SEL_CAP 6144   // finalize LDS candidate capacity

__device__ __forceinline__ unsigned orderKey(float f) {
  unsigned b = __float_as_uint(f);
  return (b & 0x80000000u) ? ~b : (b | 0x80000000u);
}
__device__ __forceinline__ float unorderKey(unsigned u) {
  unsigned b = (u & 0x80000000u) ? (u ^ 0x80000000u) : ~u;
  return __uint_as_float(b);
}

template <int N>
__device__ __forceinline__ void waitAsync() {
#if __has_builtin(__builtin_amdgcn_s_wait_asynccnt)
  __builtin_amdgcn_s_wait_asynccnt(N);
#else
  asm volatile("s_wait_asynccnt %0" :: "i"(N) : "memory");
#endif
}

// Issue exactly LPT async-DMA b128 loads per thread for one 3x18x80 halo tile.
// Border coords are clamped: replicated border values are already inside the
// 3x3 window, so this is exactly equivalent to -inf SAME padding for max-pool.
__device__ __forceinline__ void loadTileAsync(float* __restrict__ buf,
                                              unsigned long long gbase,
                                              int y, int tx0, int t) {
#pragma unroll
  for (int it = 0; it < LPT; ++it) {
    int i = t + it * THREADS;
    if (i >= LANELOADS) i = LANELOADS - 1;  // pad: harmless duplicate load
    int r = i / (JJ * (CC / 4));
    int rem = i - r * (JJ * (CC / 4));
    int j = rem / (CC / 4);
    int c4 = rem - j * (CC / 4);
    int yy = y - 1 + r;  yy = yy < 0 ? 0 : (yy > HH - 1 ? HH - 1 : yy);
    int xx = tx0 - 1 + j; xx = xx < 0 ? 0 : (xx > WW - 1 ? WW - 1 : xx);
    unsigned goff = (unsigned)(((yy * WW + xx) * CC + c4 * 4) * 4);
    unsigned loff = (unsigned)(unsigned long long)(const void*)&buf[(r * JJ + j) * CC + c4 * 4];
    asm volatile("global_load_async_to_lds_b128 %0, %1, %2"
                 :: "v"(loff), "v"(goff), "s"(gbase) : "memory");
  }
}

// One sweep over the heatmap. Each block: one half-row (16 tiles), async-DMA
// double-buffered through LDS.
// MODE 0: histogram of top-12 orderable bits           -> ghist
// MODE 1: histogram of bits [19:8] within bin state[0] -> ghist
// MODE 2: compact candidates with (u>>8) >= state[2]   -> cand
template <int MODE>
__global__ __launch_bounds__(THREADS) void nms_pass(
    const float* __restrict__ hmap, unsigned* __restrict__ ghist,
    const unsigned* __restrict__ state, unsigned* __restrict__ candCount,
    unsigned long long* __restrict__ cand, unsigned candCap) {
  __shared__ float buf[2][3 * JJ * CC];   // 2 x 17280 B double buffer
  __shared__ unsigned lhist[NBIN];        // 16384 B
  const int t = threadIdx.x;
  const int y = (int)blockIdx.x >> 1;
  const int xh = ((int)blockIdx.x & 1) * (TILES * TX);  // 0 or 256
  const unsigned long long gbase = (unsigned long long)hmap;

  if (MODE != 2)
    for (int b = t; b < NBIN; b += THREADS) lhist[b] = 0u;

  unsigned s0 = 0, s1 = 0;
  if (MODE == 1) s0 = state[0];   // bin B1
  if (MODE == 2) s1 = state[2];   // 24-bit threshold

  loadTileAsync(buf[0], gbase, y, xh, t);

  for (int k = 0; k < TILES; ++k) {
    if (k + 1 < TILES) {
      loadTileAsync(buf[(k + 1) & 1], gbase, y, xh + (k + 1) * TX, t);
      waitAsync<LPT>();   // loads complete in order: tile k is now in LDS
    } else {
      waitAsync<0>();
    }
    __syncthreads();

    const float* tile = buf[k & 1];
    const int tx0 = xh + k * TX;
#pragma unroll
    for (int e5 = 0; e5 < (TX * CC) / THREADS; ++e5) {
      int e = t + e5 * THREADS;
      int xi = e / CC;
      int c = e - xi * CC;
      float v = tile[(JJ + xi + 1) * CC + c];  // center (row 1, col xi+1)
      float m = v;
#pragma unroll
      for (int r = 0; r < 3; ++r)
#pragma unroll
        for (int dj = 0; dj < 3; ++dj)
          m = fmaxf(m, tile[(r * JJ + xi + dj) * CC + c]);
      bool keep = (v == m) && (v > 0.1f);
      float masked = keep ? v : -1.0f;
      unsigned u = orderKey(masked);
      if (MODE == 0) {
        atomicAdd(&lhist[u >> 20], 1u);
      } else if (MODE == 1) {
        if ((u >> 20) == s0) atomicAdd(&lhist[(u >> 8) & 0xFFFu], 1u);
      } else {
        if ((u >> 8) >= s1) {
          unsigned flat = (unsigned)((y * WW + (tx0 + xi)) * CC + c);
          unsigned pos = atomicAdd(candCount, 1u);
          if (pos < candCap)
            cand[pos] = (((unsigned long long)u) << 32) | (unsigned)(~flat);
        }
      }
    }
    __syncthreads();  // tile k consumed; its buffer may be overwritten next iter
  }

  if (MODE != 2) {
    for (int b = t; b < NBIN; b += THREADS) {
      unsigned h = lhist[b];
      if (h) atomicAdd(&ghist[b], h);
    }
  }
}

// Parallel suffix-scan threshold search over a 4096-bin histogram.
// phase 0: target=TOPK        -> state[0]=bin B1, state[1]=count strictly above
// phase 1: target=TOPK-state[1] -> state[2]=(B1<<12)|B2  (24-bit prefix thr)
__global__ __launch_bounds__(THREADS) void select_kernel(
    const unsigned* __restrict__ hist, unsigned* __restrict__ state, int phase) {
  __shared__ unsigned csum[THREADS];
  __shared__ unsigned sufAfter[THREADS];
  const int t = threadIdx.x;
  const unsigned target = (phase == 0) ? (unsigned)TOPK : (TOPK - state[1]);

  unsigned loc[NBIN / THREADS];  // 16 bins per thread
  unsigned s = 0;
#pragma unroll
  for (int k = 0; k < NBIN / THREADS; ++k) {
    loc[k] = hist[t * (NBIN / THREADS) + k];
    s += loc[k];
  }
  csum[t] = s;
  __syncthreads();
  if (t == 0) {
    unsigned acc = 0;
    for (int i = THREADS - 1; i >= 0; --i) { sufAfter[i] = acc; acc += csum[i]; }
  }
  __syncthreads();

  // walk this thread's 16 bins from the top; unique global crossing point
  unsigned cum = sufAfter[t];
  for (int k = NBIN / THREADS - 1; k >= 0; --k) {
    unsigned h = loc[k];
    if (cum < target && cum + h >= target) {
      unsigned b = (unsigned)(t * (NBIN / THREADS) + k);
      if (phase == 0) { state[0] = b; state[1] = cum; }
      else            { state[2] = (state[0] << 12) | b; }
    }
    cum += h;
  }
}

// Rank-based exact top-128 (sorted desc, tie -> smaller flat index), then emit.
__global__ __launch_bounds__(THREADS) void finalize_kernel(
    const unsigned long long* __restrict__ cand,
    const unsigned* __restrict__ candCount, const float* __restrict__ rreg,
    const float* __restrict__ bbox, float* __restrict__ out, unsigned candCap) {
  __shared__ unsigned long long lc[SEL_CAP];
  __shared__ unsigned long long sel[TOPK];
  const int t = threadIdx.x;
  unsigned n = *candCount;
  if (n > candCap) n = candCap;
  if (n > SEL_CAP) n = SEL_CAP;

  for (unsigned i = (unsigned)t; i < n; i += THREADS) lc[i] = cand[i];
  if (t < TOPK) sel[t] = 0ull;
  __syncthreads();

  for (unsigned i = (unsigned)t; i < n; i += THREADS) {
    unsigned long long k = lc[i];
    unsigned rank = 0;
    for (unsigned j = 0; j < n; ++j) rank += (lc[j] > k);  // keys unique
    if (rank < TOPK) sel[rank] = k;
  }
  __syncthreads();

  if (t < TOPK) {
    unsigned long long k = sel[t];
    unsigned u = (unsigned)(k >> 32);
    unsigned flat = ~((unsigned)(k & 0xFFFFFFFFull));
    float score = unorderKey(u);
    unsigned cls = flat % CC;
    unsigned pix = flat / CC;
    unsigned x = pix % WW;
    unsigned yv = pix / WW;
    float re = rreg[(size_t)pix * (2 * CC) + 2 * cls];       // even channel
    float ro = rreg[(size_t)pix * (2 * CC) + 2 * cls + 1];   // odd channel
    float cx = rintf(((float)x + re) * 4.0f);   // RTE == jnp.round
    float cy = rintf(((float)yv + ro) * 4.0f);
    float bw = bbox[(size_t)pix * 2 + 0] * 4.0f;
    float bh = bbox[(size_t)pix * 2 + 1] * 4.0f;
    out[2 * t + 0] = cx;                         // centroids (x, y)
    out[2 * t + 1] = cy;
    out[2 * TOPK + 2 * t + 0] = bw;              // box_params
    out[2 * TOPK + 2 * t + 1] = bh;
    out[4 * TOPK + t] = (float)cls;              // types
    out[5 * TOPK + t] = score;                   // scores
  }
}

extern "C" void kernel_launch(void* const* d_in, const int* in_sizes, int n_in,
                              void* d_out, int out_size, void* d_ws,
                              size_t ws_size, hipStream_t stream) {
  const float* hmap = (const float*)d_in[0];
  const float* rreg = (const float*)d_in[1];
  const float* bbox = (const float*)d_in[2];
  float* out = (float*)d_out;

  unsigned* ws = (unsigned*)d_ws;
  unsigned* hist1 = ws;                 // [0 .. 4095]
  unsigned* hist2 = ws + NBIN;          // [4096 .. 8191]
  unsigned* state = ws + 2 * NBIN;      // [+0]=B1 [+1]=cntAbove [+2]=thr24 [+3]=candCount
  unsigned* candCount = state + 3;
  unsigned long long* cand = (unsigned long long*)(ws + 2 * NBIN + 8);  // 8B aligned
  size_t headBytes = (size_t)(2 * NBIN + 8) * sizeof(unsigned);

  unsigned candCap = 0;
  if (ws_size > headBytes) {
    size_t avail = (ws_size - headBytes) / sizeof(unsigned long long);
    candCap = avail > CAND_MAX ? CAND_MAX : (unsigned)avail;
  }

  hipMemsetAsync(d_ws, 0, headBytes, stream);

  dim3 grid(HH * 2);  // 1024 blocks: one half-row (16 async-pipelined tiles) each
  nms_pass<0><<<grid, THREADS, 0, stream>>>(hmap, hist1, state, candCount, cand, candCap);
  select_kernel<<<1, THREADS, 0, stream>>>(hist1, state, 0);
  nms_pass<1><<<grid, THREADS, 0, stream>>>(hmap, hist2, state, candCount, cand, candCap);
  select_kernel<<<1, THREADS, 0, stream>>>(hist2, state, 1);
  nms_pass<2><<<grid, THREADS, 0, stream>>>(hmap, hist2, state, candCount, cand, candCap);
  finalize_kernel<<<1, THREADS, 0, stream>>>(cand, candCount, rreg, bbox, out, candCap);
}